// SpectralLinear_16647293239970
// MI455X (gfx1250) — compile-verified
//
#include <hip/hip_runtime.h>

typedef __attribute__((ext_vector_type(2))) float v2f;
typedef __attribute__((ext_vector_type(8))) float v8f;

#define DIM   2048
#define BATCH 4096
#define NR    32

__device__ __forceinline__ v8f wmma4(v2f a, v2f b, v8f c) {
    // V_WMMA_F32_16X16X4_F32: D = A(16x4) * B(4x16) + C(16x16), all f32.
    return __builtin_amdgcn_wmma_f32_16x16x4_f32(
        /*neg_a=*/false, a, /*neg_b=*/false, b,
        /*c_mod=*/(short)0, c, /*reuse_a=*/false, /*reuse_b=*/false);
}

// ---------------------------------------------------------------------------
// prep_rows: mask the reflector banks, reverse U's order, compute beta, Sig.
//   blocks 0..31  : Vm[i]     = V[i]      masked at i ;  betaV[i]
//   blocks 32..63 : Um_rev[j] = U[31-j]   masked at 31-j; betaU[j]
//   block  64     : sig[d] = 2*0.01*(sigmoid(p[d]) - 0.5) + 1.0
// ---------------------------------------------------------------------------
__global__ void prep_rows(const float* __restrict__ U, const float* __restrict__ V,
                          const float* __restrict__ p,
                          float* __restrict__ Vm, float* __restrict__ Um_rev,
                          float* __restrict__ betaV, float* __restrict__ betaU,
                          float* __restrict__ sig) {
    const int b = blockIdx.x, t = threadIdx.x;
    __shared__ float red[256];
    if (b < 64) {
        const float* src; float* dst; float* betaDst; int mi;
        if (b < 32) { src = V + b * DIM; dst = Vm + b * DIM; betaDst = betaV + b; mi = b; }
        else        { int j = b - 32; int i = 31 - j;
                      src = U + i * DIM; dst = Um_rev + j * DIM; betaDst = betaU + j; mi = i; }
        float s = 0.0f;
        for (int d = t; d < DIM; d += 256) {
            float v = (d >= mi) ? src[d] : 0.0f;
            dst[d] = v;
            s += v * v;
        }
        red[t] = s; __syncthreads();
        for (int off = 128; off > 0; off >>= 1) {
            if (t < off) red[t] += red[t + off];
            __syncthreads();
        }
        if (t == 0) *betaDst = 2.0f / red[0];
    } else {
        for (int d = t; d < DIM; d += 256) {
            float sg = 1.0f / (1.0f + __expf(-p[d]));
            sig[d] = 2.0f * 0.01f * (sg - 0.5f) + 1.0f;
        }
    }
}

// ---------------------------------------------------------------------------
// prep_T: Gram matrix G = W W^T (32x32, K=2048) then compact-WY T recurrence:
//   T_{k+1} = [[T_k, -beta_k * T_k * g(:,k)], [0, beta_k]]
// One block of 1024 threads; thread t = i*32+j owns G[i][j].
// ---------------------------------------------------------------------------
__global__ void prep_T(const float* __restrict__ W, const float* __restrict__ beta,
                       float* __restrict__ T) {
    __shared__ float g[NR][NR];
    __shared__ float Tl[NR][NR];
    const int t = threadIdx.x;
    const int i = t >> 5, j = t & 31;
    const float* wi = W + i * DIM;
    const float* wj = W + j * DIM;
    float s = 0.0f;
    for (int d = 0; d < DIM; d += 4) {
        s += wi[d] * wj[d] + wi[d + 1] * wj[d + 1]
           + wi[d + 2] * wj[d + 2] + wi[d + 3] * wj[d + 3];
    }
    g[i][j]  = s;
    Tl[i][j] = 0.0f;
    __syncthreads();
    for (int c = 0; c < NR; ++c) {
        if (t == c) {
            Tl[c][c] = beta[c];
        } else if (t < c) {
            float acc = 0.0f;
            for (int m = t; m < c; ++m) acc += Tl[t][m] * g[m][c];
            Tl[t][c] = -beta[c] * acc;
        }
        __syncthreads();
    }
    T[i * NR + j] = Tl[i][j];
}

// ---------------------------------------------------------------------------
// gemm_xwT: A = X * W^T.  M=4096, N=32, K=2048, f32 WMMA 16x16x4.
// Block = 128 rows (8 waves x 16 rows); each wave holds two 16x16 accumulators
// covering all 32 output columns, K-loop in steps of 4.
// A-frag: lane(0-15)=M, half selects K pair {k,k+1} vs {k+2,k+3} (ISA 16x4 layout).
// B-frag: B[k][n] = W[n][k] -> contiguous float2 along W's row.
// ---------------------------------------------------------------------------
__global__ void gemm_xwT(const float* __restrict__ X, const float* __restrict__ W,
                         float* __restrict__ A) {
    const int w    = threadIdx.x >> 5;
    const int lane = threadIdx.x & 31;
    const int half = lane >> 4;
    const int lq   = lane & 15;
    const int m0   = blockIdx.x * 128 + w * 16;

    const float* xrow = X + (m0 + lq) * DIM + 2 * half;
    const float* w0   = W + lq * DIM + 2 * half;
    const float* w1   = W + (lq + 16) * DIM + 2 * half;

    v8f acc0 = {}; v8f acc1 = {};
    for (int k = 0; k < DIM; k += 4) {
        __builtin_prefetch(xrow + k + 256, 0, 0);   // global_prefetch_b8, stream X
        v2f a  = *(const v2f*)(xrow + k);
        v2f b0 = *(const v2f*)(w0 + k);
        v2f b1 = *(const v2f*)(w1 + k);
        acc0 = wmma4(a, b0, acc0);
        acc1 = wmma4(a, b1, acc1);
    }
#pragma unroll
    for (int r = 0; r < 8; ++r) {
        const int row = m0 + r + 8 * half;          // C/D layout: VGPR r, half -> M
        A[row * NR + lq]      = acc0[r];
        A[row * NR + 16 + lq] = acc1[r];
    }
}

// ---------------------------------------------------------------------------
// update: out = (Xin - (A*T)*W) op vec    (op: mode=0 -> *sig , mode=1 -> +bias)
// Block tile 128x128; LDS stages T (4KB), A-tile (16KB), Z=A*T (16KB), W-tile (16KB).
// Each wave: 16 rows x 128 cols = 8 accumulators, K=32 -> 8 WMMA steps each.
// ---------------------------------------------------------------------------
__global__ void update(const float* __restrict__ Xin, const float* __restrict__ A,
                       const float* __restrict__ T, const float* __restrict__ W,
                       const float* __restrict__ vec, int mode,
                       float* __restrict__ out) {
    __shared__ float Tl[NR * NR];
    __shared__ float At[128 * NR];
    __shared__ float Z [128 * NR];
    __shared__ float Wt[NR * 128];

    const int tid = threadIdx.x;
    const int m0  = blockIdx.x * 128;
    const int n0  = blockIdx.y * 128;

    for (int idx = tid; idx < NR * NR; idx += 256) Tl[idx] = T[idx];
    for (int idx = tid; idx < 128 * NR; idx += 256) At[idx] = A[m0 * NR + idx];
    for (int idx = tid; idx < NR * 128; idx += 256) {
        const int k = idx >> 7, c = idx & 127;
        Wt[idx] = W[k * DIM + n0 + c];
    }
    __syncthreads();

    // Z = At * Tl  (128x32 * 32x32)
    for (int idx = tid; idx < 128 * NR; idx += 256) {
        const int r = idx >> 5, k = idx & 31;
        float s = 0.0f;
#pragma unroll
        for (int i = 0; i < NR; ++i) s += At[r * NR + i] * Tl[i * NR + k];
        Z[idx] = s;
    }
    __syncthreads();

    const int w    = tid >> 5;
    const int lane = tid & 31;
    const int half = lane >> 4;
    const int lq   = lane & 15;
    const int mr   = w * 16;

    v8f acc[8] = {};
    for (int k0 = 0; k0 < NR; k0 += 4) {
        const int ka = k0 + 2 * half;
        v2f a = *(const v2f*)&Z[(mr + lq) * NR + ka];
#pragma unroll
        for (int c = 0; c < 8; ++c) {
            const int n = c * 16 + lq;
            v2f b;
            b.x = Wt[ka * 128 + n];
            b.y = Wt[(ka + 1) * 128 + n];
            acc[c] = wmma4(a, b, acc[c]);
        }
    }

#pragma unroll
    for (int c = 0; c < 8; ++c) {
        const int gcol = n0 + c * 16 + lq;
        const float vv = vec[gcol];
#pragma unroll
        for (int r = 0; r < 8; ++r) {
            const int grow = m0 + mr + r + 8 * half;
            float val = Xin[grow * DIM + gcol] - acc[c][r];
            val = mode ? (val + vv) : (val * vv);
            out[grow * DIM + gcol] = val;
        }
    }
}

// ---------------------------------------------------------------------------
extern "C" void kernel_launch(void* const* d_in, const int* in_sizes, int n_in,
                              void* d_out, int out_size, void* d_ws, size_t ws_size,
                              hipStream_t stream) {
    (void)in_sizes; (void)n_in; (void)out_size; (void)ws_size;
    const float* x    = (const float*)d_in[0];   // [4096,2048]
    const float* U    = (const float*)d_in[1];   // [32,2048]
    const float* V    = (const float*)d_in[2];   // [32,2048]
    const float* p    = (const float*)d_in[3];   // [2048]
    const float* bias = (const float*)d_in[4];   // [2048]
    float* out = (float*)d_out;

    float* ws    = (float*)d_ws;
    float* Vm    = ws;                 // 32*2048 = 65536
    float* Um    = ws + 65536;         // 32*2048
    float* sig   = ws + 131072;        // 2048
    float* betaV = ws + 133120;        // 32
    float* betaU = ws + 133152;        // 32
    float* T_V   = ws + 133184;        // 1024
    float* T_U   = ws + 134208;        // 1024
    float* Abuf  = ws + 135232;        // 4096*32 = 131072

    prep_rows<<<65, 256, 0, stream>>>(U, V, p, Vm, Um, betaV, betaU, sig);
    prep_T<<<1, 1024, 0, stream>>>(Vm, betaV, T_V);
    prep_T<<<1, 1024, 0, stream>>>(Um, betaU, T_U);

    // Stage 1: X1 = (x - (x Vm^T) T_V Vm) * sig   -> d_out
    gemm_xwT<<<32, 256, 0, stream>>>(x, Vm, Abuf);
    update<<<dim3(32, 16), 256, 0, stream>>>(x, Abuf, T_V, Vm, sig, 0, out);

    // Stage 2: out = (X1 - (X1 Um^T) T_U Um) + bias   (in place on d_out)
    gemm_xwT<<<32, 256, 0, stream>>>(out, Um, Abuf);
    update<<<dim3(32, 16), 256, 0, stream>>>(out, Abuf, T_U, Um, bias, 1, out);
}